// SiameseTangentLayer_8031588844112
// MI455X (gfx1250) — compile-verified
//
#include <hip/hip_runtime.h>
#include <hip/hip_bf16.h>
#include <math.h>

typedef __attribute__((ext_vector_type(2))) float v2f;
typedef __attribute__((ext_vector_type(8))) float v8f;

#define B_SZ 1024
#define P_SZ 200
#define D_SZ 512
#define S_SZ 64
#define KC   8          // K-rows of U staged per async chunk (8 x 64 f32 = 2KB)

// Issue one 2KB U-chunk (KC rows x 64 f32) into LDS: 4 x b128 per lane,
// tracked on ASYNCcnt. INST_OFFSET is added to BOTH the LDS and global
// address per the ISA, so one base register pair serves all four issues.
__device__ __forceinline__ void issue_u_chunk(const float* gsrc, unsigned lds_byte,
                                              int lane) {
    const char* g = (const char*)gsrc + lane * 16;
    unsigned    d = lds_byte + lane * 16;
    asm volatile("global_load_async_to_lds_b128 %0, %1, off"             :: "v"(d), "v"(g) : "memory");
    asm volatile("global_load_async_to_lds_b128 %0, %1, off offset:512"  :: "v"(d), "v"(g) : "memory");
    asm volatile("global_load_async_to_lds_b128 %0, %1, off offset:1024" :: "v"(d), "v"(g) : "memory");
    asm volatile("global_load_async_to_lds_b128 %0, %1, off offset:1536" :: "v"(d), "v"(g) : "memory");
}

// ---------------------------------------------------------------------------
// Kernel 1: Gram matrices  G_p = U_p^T U_p   (P x 64 x 64), K = D.
// ---------------------------------------------------------------------------
__global__ __launch_bounds__(128) void gram_kernel(const float* __restrict__ U,
                                                   float* __restrict__ G) {
    const int p    = blockIdx.x;
    const int lane = threadIdx.x & 31;
    const int wave = threadIdx.x >> 5;      // 0..3 -> row strip
    const int half = lane >> 4;
    const int ml   = lane & 15;
    const int s0   = wave * 16;

    const float* Up = U + (size_t)p * D_SZ * S_SZ;   // [D][S]

    v8f c0 = {}, c1 = {}, c2 = {}, c3 = {};
    for (int d0 = 0; d0 < D_SZ; d0 += 4) {
        const int kv = d0 + 2 * half;       // A/B layout: v0 = K0|K2, v1 = K1|K3
        const float* uk0 = Up + (size_t)kv * S_SZ;
        const float* uk1 = uk0 + S_SZ;
        __builtin_prefetch(uk0 + 4 * S_SZ + ml, 0, 1);

        v2f a;
        a.x = uk0[s0 + ml];
        a.y = uk1[s0 + ml];
        v2f b0, b1, b2, b3;
        b0.x = uk0[ 0 + ml]; b0.y = uk1[ 0 + ml];
        b1.x = uk0[16 + ml]; b1.y = uk1[16 + ml];
        b2.x = uk0[32 + ml]; b2.y = uk1[32 + ml];
        b3.x = uk0[48 + ml]; b3.y = uk1[48 + ml];

        c0 = __builtin_amdgcn_wmma_f32_16x16x4_f32(false, a, false, b0, (short)0, c0, false, false);
        c1 = __builtin_amdgcn_wmma_f32_16x16x4_f32(false, a, false, b1, (short)0, c1, false, false);
        c2 = __builtin_amdgcn_wmma_f32_16x16x4_f32(false, a, false, b2, (short)0, c2, false, false);
        c3 = __builtin_amdgcn_wmma_f32_16x16x4_f32(false, a, false, b3, (short)0, c3, false, false);
    }

    float* Gp = G + (size_t)p * S_SZ * S_SZ;
#pragma unroll
    for (int i = 0; i < 8; ++i) {
        const int row = s0 + half * 8 + i;
        Gp[row * S_SZ +  0 + ml] = c0[i];
        Gp[row * S_SZ + 16 + ml] = c1[i];
        Gp[row * S_SZ + 32 + ml] = c2[i];
        Gp[row * S_SZ + 48 + ml] = c3[i];
    }
}

// ---------------------------------------------------------------------------
// Kernel 2: per 16-row batch tile x 8 prototypes (one wave each):
//   pass1: t = (x - c_p) @ U_p   (K=512; U double-buffered in LDS via
//          global_load_async_to_lds_b128 + s_wait_asynccnt)
//   pass2: h = t @ G_p           (K=64)
//   out   = sqrt(||diff||^2 - 2 t.t + t.h)  (register shuffle reduction)
// Static LDS: 32KB x-tile (reused for t staging) + 32KB U buffers = 64KB.
// ---------------------------------------------------------------------------
__global__ __launch_bounds__(256) void tangent_kernel(
    const float* __restrict__ x, const float* __restrict__ protos,
    const float* __restrict__ U, const float* __restrict__ G,
    float* __restrict__ out) {
    __shared__ float xs[16 * D_SZ];                        // 32KB; reused as t[8][16][64]
    __shared__ __align__(16) float us[8 * 2 * KC * S_SZ];  // 32KB: U double buffers

    const int lane  = threadIdx.x & 31;
    const int wave  = threadIdx.x >> 5;
    const int half  = lane >> 4;
    const int ml    = lane & 15;
    const int brow0 = blockIdx.x * 16;
    const int p     = blockIdx.y * 8 + wave;

    // Cooperative coalesced load of x tile [16][512].
    {
        const float4* xg = (const float4*)(x + (size_t)brow0 * D_SZ);
        float4*       xl = (float4*)xs;
        for (int i = threadIdx.x; i < 16 * D_SZ / 4; i += 256)
            xl[i] = xg[i];
    }
    __syncthreads();

    const float* Up = U + (size_t)p * D_SZ * S_SZ;   // [D][S]
    const float* pp = protos + (size_t)p * D_SZ;
    const float* Gp = G + (size_t)p * S_SZ * S_SZ;

    float*         usw    = us + wave * 2 * KC * S_SZ;
    const unsigned ubyte0 = (unsigned)(uintptr_t)usw;           // LDS byte offset (low 32b)
    const unsigned ubyte1 = ubyte0 + KC * S_SZ * 4;

    // ---- pass 1: t = diff @ U_p, plus per-row ||diff||^2 ----
    v8f t0 = {}, t1 = {}, t2 = {}, t3 = {};
    float dn2 = 0.f;

    issue_u_chunk(Up, ubyte0, lane);                            // prime buffer 0
    for (int kb = 0, buf = 0; kb < D_SZ; kb += KC, buf ^= 1) {
        if (kb + KC < D_SZ) {
            issue_u_chunk(Up + (size_t)(kb + KC) * S_SZ,
                          buf ? ubyte0 : ubyte1, lane);
            asm volatile("s_wait_asynccnt 0x4" ::: "memory");   // current chunk landed
        } else {
            asm volatile("s_wait_asynccnt 0x0" ::: "memory");
        }
        const float* ub = usw + (buf ? KC * S_SZ : 0);

#pragma unroll
        for (int j = 0; j < KC; j += 4) {
            const int kv = kb + j + 2 * half;   // global K index for this lane-half
            const int kl = j + 2 * half;        // row within the LDS chunk

            const float p0 = pp[kv], p1 = pp[kv + 1];
            v2f a;
            a.x = xs[ml * D_SZ + kv]     - p0;
            a.y = xs[ml * D_SZ + kv + 1] - p1;
            dn2 += a.x * a.x + a.y * a.y;

            const float* u0 = ub + kl * S_SZ;
            const float* u1 = u0 + S_SZ;
            v2f b0, b1, b2, b3;
            b0.x = u0[ 0 + ml]; b0.y = u1[ 0 + ml];
            b1.x = u0[16 + ml]; b1.y = u1[16 + ml];
            b2.x = u0[32 + ml]; b2.y = u1[32 + ml];
            b3.x = u0[48 + ml]; b3.y = u1[48 + ml];

            t0 = __builtin_amdgcn_wmma_f32_16x16x4_f32(false, a, false, b0, (short)0, t0, false, false);
            t1 = __builtin_amdgcn_wmma_f32_16x16x4_f32(false, a, false, b1, (short)0, t1, false, false);
            t2 = __builtin_amdgcn_wmma_f32_16x16x4_f32(false, a, false, b2, (short)0, t2, false, false);
            t3 = __builtin_amdgcn_wmma_f32_16x16x4_f32(false, a, false, b3, (short)0, t3, false, false);
        }
    }
    dn2 += __shfl_xor(dn2, 16, 32);             // rows l and l^16 are the same m

    // ---- spill t into the (now idle) x-tile LDS region for pass-2 A feed ----
    __syncthreads();                            // everyone done reading xs
    float* tw = xs + wave * 16 * S_SZ;
#pragma unroll
    for (int i = 0; i < 8; ++i) {
        const int row = half * 8 + i;
        tw[row * S_SZ +  0 + ml] = t0[i];
        tw[row * S_SZ + 16 + ml] = t1[i];
        tw[row * S_SZ + 32 + ml] = t2[i];
        tw[row * S_SZ + 48 + ml] = t3[i];
    }
    __syncthreads();

    // ---- pass 2: h = t @ G_p  (K = 64) ----
    v8f h0 = {}, h1 = {}, h2 = {}, h3 = {};
    for (int s0 = 0; s0 < S_SZ; s0 += 4) {
        const int kv = s0 + 2 * half;
        v2f a;
        a.x = tw[ml * S_SZ + kv];
        a.y = tw[ml * S_SZ + kv + 1];

        const float* gk0 = Gp + (size_t)kv * S_SZ;
        const float* gk1 = gk0 + S_SZ;
        v2f b0, b1, b2, b3;
        b0.x = gk0[ 0 + ml]; b0.y = gk1[ 0 + ml];
        b1.x = gk0[16 + ml]; b1.y = gk1[16 + ml];
        b2.x = gk0[32 + ml]; b2.y = gk1[32 + ml];
        b3.x = gk0[48 + ml]; b3.y = gk1[48 + ml];

        h0 = __builtin_amdgcn_wmma_f32_16x16x4_f32(false, a, false, b0, (short)0, h0, false, false);
        h1 = __builtin_amdgcn_wmma_f32_16x16x4_f32(false, a, false, b1, (short)0, h1, false, false);
        h2 = __builtin_amdgcn_wmma_f32_16x16x4_f32(false, a, false, b2, (short)0, h2, false, false);
        h3 = __builtin_amdgcn_wmma_f32_16x16x4_f32(false, a, false, b3, (short)0, h3, false, false);
    }

    // ---- epilogue, all in registers:  out^2[m] = dn2 + sum_s t*(h - 2t) ----
    // C layout: VGPR i holds row (half*8+i); lanes of a half span 16 s-values,
    // the 4 accumulators span the 4 s-tiles. Reduce over s with xor shuffles
    // that stay inside each 16-lane half.
    float part[8];
#pragma unroll
    for (int i = 0; i < 8; ++i) {
        float v = t0[i] * (h0[i] - 2.f * t0[i])
                + t1[i] * (h1[i] - 2.f * t1[i])
                + t2[i] * (h2[i] - 2.f * t2[i])
                + t3[i] * (h3[i] - 2.f * t3[i]);
        v += __shfl_xor(v, 1, 32);
        v += __shfl_xor(v, 2, 32);
        v += __shfl_xor(v, 4, 32);
        v += __shfl_xor(v, 8, 32);
        part[i] = v;                 // uniform across the 16 lanes of this half
    }

    // Writers: lanes 0..7 -> rows 0..7 (idx=lane); lanes 24..31 -> rows 8..15
    // (idx=lane-24). Both have dn2 for exactly that row (ml == row).
    int row = -1, idx = 0;
    if (lane < 8)       { row = lane;      idx = lane;      }
    else if (lane >= 24){ row = lane - 16; idx = lane - 24; }
    if (row >= 0) {
        float sel = 0.f;
#pragma unroll
        for (int i = 0; i < 8; ++i) sel = (idx == i) ? part[i] : sel;
        const float o2 = dn2 + sel;
        out[(size_t)(brow0 + row) * P_SZ + p] = sqrtf(fmaxf(o2, 0.f));
    }
}

// ---------------------------------------------------------------------------
extern "C" void kernel_launch(void* const* d_in, const int* in_sizes, int n_in,
                              void* d_out, int out_size, void* d_ws, size_t ws_size,
                              hipStream_t stream) {
    const float* x      = (const float*)d_in[0];   // [B, D]
    const float* protos = (const float*)d_in[1];   // [P, D]
    const float* U      = (const float*)d_in[2];   // [P, D, S]
    float*       out    = (float*)d_out;           // [B, P]
    float*       G      = (float*)d_ws;            // [P, S, S] scratch (3.28 MB)

    gram_kernel<<<dim3(P_SZ), dim3(128), 0, stream>>>(U, G);

    dim3 grid(B_SZ / 16, P_SZ / 8);                // 64 x 25 blocks
    tangent_kernel<<<grid, dim3(256), 0, stream>>>(x, protos, U, G, out);
}